// EdgeFeatures_53085795779045
// MI455X (gfx1250) — compile-verified
//
#include <hip/hip_runtime.h>
#include <hip/hip_bf16.h>

// Problem constants (match reference)
#define Bc    4
#define Nc    2048
#define KNN   30
#define EF    128
#define KIN   39        // 16 PE + 16 RBF + 3 dU + 4 quat
#define KPAD  40        // pad K to multiple of 4 for wmma f32 16x16x4
#define ROWS_TOTAL (Bc * Nc * KNN)   // 245760 edge rows

typedef float v2f __attribute__((ext_vector_type(2)));
typedef float v8f __attribute__((ext_vector_type(8)));

__device__ __forceinline__ float sgnf(float x) {
    return (x > 0.f) ? 1.f : ((x < 0.f) ? -1.f : 0.f);
}
__device__ __forceinline__ void l2norm3(const float* v, float* o) {
    float n = sqrtf(v[0]*v[0] + v[1]*v[1] + v[2]*v[2]);
    n = fmaxf(n, 1e-12f);
    float inv = 1.f / n;
    o[0] = v[0]*inv; o[1] = v[1]*inv; o[2] = v[2]*inv;
}
__device__ __forceinline__ void cross3(const float* a, const float* b, float* o) {
    o[0] = a[1]*b[2] - a[2]*b[1];
    o[1] = a[2]*b[0] - a[0]*b[2];
    o[2] = a[0]*b[1] - a[1]*b[0];
}

// ---------------------------------------------------------------------------
// Kernel 1: local orientation frames O9[b][i][9] = rows {o1, n2, o1 x n2}.
// Valid for i in [1, N-3]; rows 0, N-2, N-1 are zero (the jnp.pad).
// ---------------------------------------------------------------------------
__global__ void frames_kernel(const float* __restrict__ X, float* __restrict__ O9) {
    int g = blockIdx.x * blockDim.x + threadIdx.x;
    if (g >= Bc * Nc) return;
    int i = g % Nc;
    float o9[9];
#pragma unroll
    for (int t = 0; t < 9; ++t) o9[t] = 0.f;
    if (i >= 1 && i <= Nc - 3) {
        const float* xm = X + (size_t)(g - 1) * 3;   // X[b, i-1]
        const float* x0 = xm + 3;                    // X[b, i]
        const float* xp = xm + 6;                    // X[b, i+1]
        float d0[3] = { x0[0]-xm[0], x0[1]-xm[1], x0[2]-xm[2] };
        float d1[3] = { xp[0]-x0[0], xp[1]-x0[1], xp[2]-x0[2] };
        float u2[3], u1[3];
        l2norm3(d0, u2);
        l2norm3(d1, u1);
        float cr[3]; cross3(u2, u1, cr);
        float n2[3]; l2norm3(cr, n2);
        float dm[3] = { u2[0]-u1[0], u2[1]-u1[1], u2[2]-u1[2] };
        float o1[3]; l2norm3(dm, o1);
        float c2[3]; cross3(o1, n2, c2);
        o9[0]=o1[0]; o9[1]=o1[1]; o9[2]=o1[2];
        o9[3]=n2[0]; o9[4]=n2[1]; o9[5]=n2[2];
        o9[6]=c2[0]; o9[7]=c2[1]; o9[8]=c2[2];
    }
#pragma unroll
    for (int t = 0; t < 9; ++t) O9[(size_t)g * 9 + t] = o9[t];
}

// ---------------------------------------------------------------------------
// Kernel 2: one wave32 per node. Distance row in LDS, mask-adjusted with
// row max, then 30 rounds of wave-wide argmin (knockout selection) =
// jax.lax.top_k(-D) order (ascending distance, ties -> lower index).
// ---------------------------------------------------------------------------
__global__ __launch_bounds__(32) void topk_kernel(
    const float* __restrict__ X, const float* __restrict__ mask,
    int* __restrict__ wsIdx, float* __restrict__ wsDn,
    float* __restrict__ outIdxF)
{
    __shared__ float row[Nc];
    int bn = blockIdx.x;
    int b  = bn / Nc;
    int lane = threadIdx.x;

    const float* xn = X + (size_t)bn * 3;
    float xnx = xn[0], xny = xn[1], xnz = xn[2];
    float mn = mask[bn];

    float dmax = -1.f;
    for (int j = lane; j < Nc; j += 32) {
        const float* xj = X + (size_t)(b * Nc + j) * 3;
        float dx = xnx - xj[0], dy = xny - xj[1], dz = xnz - xj[2];
        float d  = sqrtf(dx*dx + dy*dy + dz*dz + 1e-6f);
        float m2 = mn * mask[b * Nc + j];
        float D  = m2 * d;
        row[j] = D;
        dmax = fmaxf(dmax, D);
    }
#pragma unroll
    for (int off = 16; off > 0; off >>= 1)
        dmax = fmaxf(dmax, __shfl_xor(dmax, off, 32));
    __syncthreads();

    // D_adjust = D + (1 - mask2d) * D_max
    for (int j = lane; j < Nc; j += 32) {
        float m2 = mn * mask[b * Nc + j];
        row[j] = row[j] + (1.f - m2) * dmax;
    }
    __syncthreads();

    for (int t = 0; t < KNN; ++t) {
        float best = 3.0e38f;
        int   bi   = Nc;
        for (int j = lane; j < Nc; j += 32) {
            float v = row[j];
            if (v < best || (v == best && j < bi)) { best = v; bi = j; }
        }
#pragma unroll
        for (int off = 16; off > 0; off >>= 1) {
            float ov = __shfl_xor(best, off, 32);
            int   oi = __shfl_xor(bi,   off, 32);
            if (ov < best || (ov == best && oi < bi)) { best = ov; bi = oi; }
        }
        if (lane == 0) {
            wsIdx[bn * KNN + t]   = bi;
            wsDn [bn * KNN + t]   = best;
            outIdxF[bn * KNN + t] = (float)bi;
            row[bi] = 3.4e38f;    // knockout
        }
        __syncthreads();
    }
}

// ---------------------------------------------------------------------------
// Kernel 3: feature assembly + (128-row x 39) @ (39 x 128) via
// V_WMMA_F32_16X16X4_F32 (10 K-steps over KPAD=40) + bias + LayerNorm.
// Block: 256 threads = 8 waves; each wave owns a 16-row x 128-col tile.
// ---------------------------------------------------------------------------
__global__ __launch_bounds__(256) void edge_kernel(
    const float* __restrict__ X,  const float* __restrict__ W,
    const float* __restrict__ bv, const float* __restrict__ gam,
    const float* __restrict__ bet,
    const int*   __restrict__ wsIdx, const float* __restrict__ wsDn,
    const float* __restrict__ O9, float* __restrict__ outE)
{
    __shared__ float sW[KPAD * EF];       // 40 x 128 (row 39 zero-padded)
    __shared__ float sA[128 * KPAD];      // 128 edge rows x 40 features
    __shared__ float sBias[EF], sGamma[EF], sBeta[EF];

    int tid = threadIdx.x;

    // Stage weights (padded) and per-channel constants.
    for (int i = tid; i < KPAD * EF; i += 256) {
        int r = i / EF, c = i % EF;
        sW[i] = (r < KIN) ? W[r * EF + c] : 0.f;
    }
    if (tid < EF) { sBias[tid] = bv[tid]; sGamma[tid] = gam[tid]; sBeta[tid] = bet[tid]; }

    // ---- Feature phase: threads 0..127 each build one edge row ----
    if (tid < 128) {
        int g = blockIdx.x * 128 + tid;
        float feat[KPAD];
#pragma unroll
        for (int c = 0; c < KPAD; ++c) feat[c] = 0.f;
        if (g < ROWS_TOTAL) {
            int b   = g / (Nc * KNN);
            int rem = g % (Nc * KNN);
            int n   = rem / KNN;
            int bn  = b * Nc + n;
            int j   = wsIdx[g];
            float dn = wsDn[g];

            // positional encodings: cos then sin, 8 each
            float d = (float)j - (float)n;
            const float lf = -0.5756462732485115f;   // -ln(10000)/16
#pragma unroll
            for (int p = 0; p < 8; ++p) {
                float fr  = expf((float)(2 * p) * lf);
                float ang = d * fr;
                feat[p]     = cosf(ang);
                feat[8 + p] = sinf(ang);
            }
            // RBF: mu = linspace(0,20,16), sigma = 1.25
#pragma unroll
            for (int i = 0; i < 16; ++i) {
                float mu = 20.f * (float)i * (1.f / 15.f);
                float t  = (dn - mu) * (1.f / 1.25f);
                feat[16 + i] = expf(-t * t);
            }
            // orientation features
            float Om[9], On[9];
#pragma unroll
            for (int t = 0; t < 9; ++t) {
                Om[t] = O9[(size_t)bn * 9 + t];
                On[t] = O9[(size_t)(b * Nc + j) * 9 + t];
            }
            float dX[3] = {
                X[(size_t)(b*Nc + j)*3 + 0] - X[(size_t)bn*3 + 0],
                X[(size_t)(b*Nc + j)*3 + 1] - X[(size_t)bn*3 + 1],
                X[(size_t)(b*Nc + j)*3 + 2] - X[(size_t)bn*3 + 2] };
            float du[3];
#pragma unroll
            for (int i = 0; i < 3; ++i)
                du[i] = Om[i*3+0]*dX[0] + Om[i*3+1]*dX[1] + Om[i*3+2]*dX[2];
            float dun[3]; l2norm3(du, dun);
            feat[32] = dun[0]; feat[33] = dun[1]; feat[34] = dun[2];

            // R = Om^T @ On
            float R[9];
#pragma unroll
            for (int i = 0; i < 3; ++i)
#pragma unroll
                for (int l = 0; l < 3; ++l)
                    R[i*3+l] = Om[0*3+i]*On[0*3+l] + Om[1*3+i]*On[1*3+l] + Om[2*3+i]*On[2*3+l];
            float Rxx = R[0], Ryy = R[4], Rzz = R[8];
            float m0 = 0.5f * sqrtf(fabsf(1.f + Rxx - Ryy - Rzz));
            float m1 = 0.5f * sqrtf(fabsf(1.f - Rxx + Ryy - Rzz));
            float m2 = 0.5f * sqrtf(fabsf(1.f - Rxx - Ryy + Rzz));
            float s0 = sgnf(R[2*3+1] - R[1*3+2]);
            float s1 = sgnf(R[0*3+2] - R[2*3+0]);
            float s2 = sgnf(R[1*3+0] - R[0*3+1]);
            float q0 = s0*m0, q1 = s1*m1, q2 = s2*m2;
            float qw = sqrtf(fmaxf(0.f, 1.f + Rxx + Ryy + Rzz)) * 0.5f;
            float qn = sqrtf(q0*q0 + q1*q1 + q2*q2 + qw*qw);
            qn = fmaxf(qn, 1e-12f);
            float qi = 1.f / qn;
            feat[35] = q0*qi; feat[36] = q1*qi; feat[37] = q2*qi; feat[38] = qw*qi;
        }
#pragma unroll
        for (int c = 0; c < KPAD; ++c) sA[tid * KPAD + c] = feat[c];
    }
    __syncthreads();

    // ---- GEMM phase: wave w -> rows [16w, 16w+16), 8 x 16-col tiles ----
    int wave = tid >> 5;
    int lane = tid & 31;
    int m    = lane & 15;            // M (for A) / N (for B) within tile
    int kOff = (lane >> 4) * 2;      // K pair: lanes 0-15 -> {0,1}, 16-31 -> {2,3}

    v8f acc[8];
    v8f z = {0.f,0.f,0.f,0.f,0.f,0.f,0.f,0.f};
#pragma unroll
    for (int nt = 0; nt < 8; ++nt) acc[nt] = z;

#pragma unroll
    for (int kk = 0; kk < KPAD / 4; ++kk) {
        v2f a;
        a.x = sA[(wave * 16 + m) * KPAD + kk * 4 + kOff];
        a.y = sA[(wave * 16 + m) * KPAD + kk * 4 + kOff + 1];
#pragma unroll
        for (int nt = 0; nt < 8; ++nt) {
            v2f bb;
            bb.x = sW[(kk * 4 + kOff    ) * EF + nt * 16 + m];
            bb.y = sW[(kk * 4 + kOff + 1) * EF + nt * 16 + m];
            acc[nt] = __builtin_amdgcn_wmma_f32_16x16x4_f32(
                false, a, false, bb, (short)0, acc[nt], false, false);
        }
    }

    // ---- Bias + LayerNorm (ddof=1) from fragments, width-16 reductions ----
    float pS[8], pQ[8];
#pragma unroll
    for (int r = 0; r < 8; ++r) { pS[r] = 0.f; pQ[r] = 0.f; }
#pragma unroll
    for (int nt = 0; nt < 8; ++nt) {
#pragma unroll
        for (int r = 0; r < 8; ++r) {
            int ncol = m + nt * 16;
            float v = acc[nt][r] + sBias[ncol];
            acc[nt][r] = v;
            pS[r] += v;
            pQ[r] += v * v;
        }
    }
#pragma unroll
    for (int off = 1; off < 16; off <<= 1) {
#pragma unroll
        for (int r = 0; r < 8; ++r) {
            pS[r] += __shfl_xor(pS[r], off, 16);
            pQ[r] += __shfl_xor(pQ[r], off, 16);
        }
    }
    int hi = lane >> 4;   // which half-wave: rows 0-7 vs 8-15 of the tile
#pragma unroll
    for (int r = 0; r < 8; ++r) {
        int  mrow = r + hi * 8;
        long grow = (long)blockIdx.x * 128 + wave * 16 + mrow;
        if (grow < ROWS_TOTAL) {
            float mu  = pS[r] * (1.f / 128.f);
            float var = (pQ[r] - 128.f * mu * mu) * (1.f / 127.f);
            float sig = sqrtf(var + 1e-6f);
            float inv = 1.f / (sig + 1e-6f);
#pragma unroll
            for (int nt = 0; nt < 8; ++nt) {
                int ncol = m + nt * 16;
                float v = acc[nt][r];
                outE[grow * EF + ncol] = sGamma[ncol] * (v - mu) * inv + sBeta[ncol];
            }
        }
    }
}

// ---------------------------------------------------------------------------
extern "C" void kernel_launch(void* const* d_in, const int* in_sizes, int n_in,
                              void* d_out, int out_size, void* d_ws, size_t ws_size,
                              hipStream_t stream) {
    const float* X    = (const float*)d_in[0];
    const float* mask = (const float*)d_in[1];
    const float* W    = (const float*)d_in[2];
    const float* bv   = (const float*)d_in[3];
    const float* gam  = (const float*)d_in[4];
    const float* bet  = (const float*)d_in[5];

    float* out     = (float*)d_out;
    float* outE    = out;                                  // (B,N,KNN,EF)
    float* outIdxF = out + (size_t)ROWS_TOTAL * EF;        // (B,N,KNN) as float

    char*  ws    = (char*)d_ws;
    int*   wsIdx = (int*)  ws;                               // ROWS_TOTAL ints
    float* wsDn  = (float*)(ws + (size_t)ROWS_TOTAL * 4);    // ROWS_TOTAL floats
    float* wsO9  = (float*)(ws + (size_t)ROWS_TOTAL * 8);    // B*N*9 floats

    frames_kernel<<<(Bc * Nc + 255) / 256, 256, 0, stream>>>(X, wsO9);
    topk_kernel<<<Bc * Nc, 32, 0, stream>>>(X, mask, wsIdx, wsDn, outIdxF);
    edge_kernel<<<(ROWS_TOTAL + 127) / 128, 256, 0, stream>>>(
        X, W, bv, gam, bet, wsIdx, wsDn, wsO9, outE);
}